// TwoBranchDH_SFNN_2637109920467
// MI455X (gfx1250) — compile-verified
//
#include <hip/hip_runtime.h>
#include <hip/hip_bf16.h>

// MI455X / gfx1250 fused two-branch SFNN.
// One block per batch element; recurrent state in registers; projections via
// V_WMMA_F32_16X16X4_F32 (fp32, K=4 -> 5 steps for K=20); W1/W2 fragments
// preloaded into registers; x chunks register-double-buffered to hide global
// latency behind the WMMA+scan phase; output reduction done as a transposed
// LDS pass (wave w reduces timestep w) instead of per-t shuffles.

typedef float v2f __attribute__((ext_vector_type(2)));
typedef float v8f __attribute__((ext_vector_type(8)));

#define B_      128
#define T_      2048
#define H_      512
#define DIN_    40
#define K1_     20      // per-branch reduction dim
#define TC_     16      // time chunk == WMMA M == number of waves
#define NWAVE_  16
#define XS_STRIDE_ 41   // 40 padded: gcd(41,64)=1 -> conflict-free A-frag reads
#define I_STRIDE_  516  // 512 padded: 8*516 % 64 == 32 -> conflict-free half-tile writes

__global__ __launch_bounds__(512)
void sfnn_fused_wmma_kernel(const float* __restrict__ x,
                            const float* __restrict__ W1, const float* __restrict__ b1,
                            const float* __restrict__ W2, const float* __restrict__ b2,
                            const float* __restrict__ Wo, const float* __restrict__ bo,
                            const float* __restrict__ tau_m,
                            const float* __restrict__ tau_n1,
                            const float* __restrict__ tau_n2,
                            float* __restrict__ out)
{
    extern __shared__ float smem[];
    float* xs   = smem;                       // [TC_][XS_STRIDE_]
    float* I1   = xs + TC_ * XS_STRIDE_;      // [TC_][I_STRIDE_]
    float* I2   = I1 + TC_ * I_STRIDE_;       // [TC_][I_STRIDE_]
    float* Cbuf = I2 + TC_ * I_STRIDE_;       // [TC_][I_STRIDE_]  woh*mem staging

    const int tid  = threadIdx.x;
    const int lane = tid & 31;
    const int wave = tid >> 5;
    const int b    = blockIdx.x;

    // wave -> (branch, 4 consecutive h-tiles)
    const int branch = wave >> 3;              // waves 0-7: branch0 (W1), 8-15: branch1 (W2)
    const int htbase = (wave & 7) * 4;         // h-tiles [htbase .. htbase+3]
    const int n      = lane & 15;              // N (h) within tile / M for A frags
    const int koff   = (lane >> 4) * 2;        // K sub-offset for hi half-wave
    const int mhi    = (lane >> 4) * 8;        // D-tile row offset for hi half-wave

    // ---- Preload B-matrix fragments (time-invariant) into registers ----
    const float* Wsrc = branch ? W2 : W1;
    v2f bf[4][5];
    #pragma unroll
    for (int j = 0; j < 4; ++j) {
        const int h = (htbase + j) * 16 + n;
        const float* wrow = Wsrc + h * K1_;
        #pragma unroll
        for (int kk = 0; kk < 5; ++kk) {
            bf[j][kk].x = wrow[4 * kk + koff];
            bf[j][kk].y = wrow[4 * kk + koff + 1];
        }
    }

    // ---- Per-thread scan constants (h == tid) ----
    const float alpha = 1.0f / (1.0f + __expf(-tau_m[tid]));
    const float beta1 = 1.0f / (1.0f + __expf(-tau_n1[tid]));
    const float beta2 = 1.0f / (1.0f + __expf(-tau_n2[tid]));
    const float b1v = b1[tid];
    const float b2v = b2[tid];
    const float woh = Wo[tid];
    const float bov = bo[0];

    float d1 = 0.0f, d2 = 0.0f, mem = 0.0f;

    const float* xb = x + (size_t)b * T_ * DIN_;

    // ---- Register double-buffer for x chunks (640 floats, <=2 per thread) ----
    const int  idx0  = tid;                       // always < 640
    const int  lds0  = (idx0 / DIN_) * XS_STRIDE_ + (idx0 % DIN_);
    const bool has1  = tid < (TC_ * DIN_ - 512);  // first 128 threads carry a 2nd elem
    const int  idx1  = tid + 512;
    const int  lds1  = (idx1 / DIN_) * XS_STRIDE_ + (idx1 % DIN_);

    float xr0 = xb[idx0];
    float xr1 = has1 ? xb[idx1] : 0.0f;

    for (int t0 = 0; t0 < T_; t0 += TC_) {
        // ---- Publish staged chunk to LDS ----
        xs[lds0] = xr0;
        if (has1) xs[lds1] = xr1;
        __syncthreads();

        // ---- Kick off next chunk's global loads (overlap with WMMA+scan) ----
        if (t0 + TC_ < T_) {
            const float* xn = xb + (t0 + TC_) * DIN_;
            xr0 = xn[idx0];
            if (has1) xr1 = xn[idx1];
            if (tid < TC_ && t0 + 2 * TC_ < T_) {
                __builtin_prefetch(xb + (t0 + 2 * TC_) * DIN_ + tid * DIN_, 0, 3);
            }
        }

        // ---- Projection GEMM: 4 tiles of [16t x 16h], K=20 in 5 WMMA steps ----
        v8f acc[4];
        #pragma unroll
        for (int j = 0; j < 4; ++j) acc[j] = (v8f){0.f,0.f,0.f,0.f,0.f,0.f,0.f,0.f};

        #pragma unroll
        for (int kk = 0; kk < 5; ++kk) {
            v2f a;
            a.x = xs[n * XS_STRIDE_ + branch * K1_ + 4 * kk + koff];
            a.y = xs[n * XS_STRIDE_ + branch * K1_ + 4 * kk + koff + 1];
            #pragma unroll
            for (int j = 0; j < 4; ++j) {
                acc[j] = __builtin_amdgcn_wmma_f32_16x16x4_f32(
                    /*neg_a=*/false, a, /*neg_b=*/false, bf[j][kk],
                    /*c_mod=*/(short)0, acc[j], /*reuse_a=*/false, /*reuse_b=*/false);
            }
        }

        // ---- Spill D tiles to LDS (rows r and r+8 per half-wave) ----
        float* Idst = branch ? I2 : I1;
        #pragma unroll
        for (int j = 0; j < 4; ++j) {
            const int hcol = (htbase + j) * 16 + n;
            #pragma unroll
            for (int r = 0; r < 8; ++r) {
                Idst[(r + mhi) * I_STRIDE_ + hcol] = acc[j][r];
            }
        }
        __syncthreads();

        // ---- Sequential scan over the 16 staged timesteps (thread = h) ----
        #pragma unroll 4
        for (int t = 0; t < TC_; ++t) {
            const float i1 = I1[t * I_STRIDE_ + tid] + b1v;
            const float i2 = I2[t * I_STRIDE_ + tid] + b2v;
            d1  = beta1 * d1 + (1.0f - beta1) * i1;
            d2  = beta2 * d2 + (1.0f - beta2) * i2;
            mem = alpha * mem + (1.0f - alpha) * (d1 + d2);
            Cbuf[t * I_STRIDE_ + tid] = woh * mem;
        }
        __syncthreads();

        // ---- Transposed output reduction: wave w reduces timestep w ----
        {
            const float* crow = Cbuf + wave * I_STRIDE_;
            float s = 0.0f;
            #pragma unroll
            for (int j = 0; j < H_ / 32; ++j) s += crow[lane + 32 * j];
            #pragma unroll
            for (int off = 16; off > 0; off >>= 1)
                s += __shfl_down(s, off, 32);
            if (lane == 0) {
                out[(size_t)b * T_ + t0 + wave] = 1.0f / (1.0f + __expf(-(s + bov)));
            }
        }
        __syncthreads();  // xs / I / Cbuf reused next chunk
    }
}

extern "C" void kernel_launch(void* const* d_in, const int* in_sizes, int n_in,
                              void* d_out, int out_size, void* d_ws, size_t ws_size,
                              hipStream_t stream) {
    (void)in_sizes; (void)n_in; (void)d_ws; (void)ws_size; (void)out_size;
    const float* x      = (const float*)d_in[0];
    const float* W1     = (const float*)d_in[1];
    const float* b1     = (const float*)d_in[2];
    const float* W2     = (const float*)d_in[3];
    const float* b2     = (const float*)d_in[4];
    const float* Wo     = (const float*)d_in[5];
    const float* bo     = (const float*)d_in[6];
    const float* tau_m  = (const float*)d_in[7];
    const float* tau_n1 = (const float*)d_in[8];
    const float* tau_n2 = (const float*)d_in[9];
    float* out = (float*)d_out;

    const size_t smem_bytes =
        (size_t)(TC_ * XS_STRIDE_ + 3 * TC_ * I_STRIDE_) * sizeof(float);

    static bool attr_set = false;
    if (!attr_set) {
        hipFuncSetAttribute((const void*)sfnn_fused_wmma_kernel,
                            hipFuncAttributeMaxDynamicSharedMemorySize,
                            (int)smem_bytes);
        attr_set = true;
    }

    sfnn_fused_wmma_kernel<<<B_, 512, smem_bytes, stream>>>(
        x, W1, b1, W2, b2, Wo, bo, tau_m, tau_n1, tau_n2, out);
}